// SimpleRNN_28578712387987
// MI455X (gfx1250) — compile-verified
//
#include <hip/hip_runtime.h>

// Elman RNN forward, fused single kernel for MI455X (gfx1250).
// h_t = relu(x_t @ W_ih^T + (b_ih+b_hh) + h_{t-1} @ W_hh^T), then
// pred = h_T @ W_out^T + b_out.
//
// One wave32 per 16-batch-row tile; weights live in VGPRs as f32 WMMA
// B-fragments, hidden state bounces through padded LDS between D-layout and
// A-layout. All math fp32 via V_WMMA_F32_16X16X4_F32 (8 K-chunks cover K=32).
//
// Scheduling: x fragments are software-pipelined one timestep ahead (loads
// for t+1 issue before step t's compute), and the input/recurrent GEMMs use
// separate accumulators so the LDS-gated recurrent chain is only 8 WMMAs.

typedef __attribute__((ext_vector_type(2))) float v2f;
typedef __attribute__((ext_vector_type(8))) float v8f;

#define RNN_H   32
#define RNN_IN  32
#define RNN_T   168
#define RNN_B   4096
#define ROWS    16          // batch rows per wave/tile
#define LSTRIDE 34          // LDS row stride in floats (even, bank-conflict-free)

__global__ __launch_bounds__(32)
void SimpleRNN_28578712387987_kernel(
    const float* __restrict__ x,        // [B, T, IN]
    const float* __restrict__ hidden,   // [1, B, H]
    const float* __restrict__ W_ih,     // [H, IN]
    const float* __restrict__ W_hh,     // [H, H]
    const float* __restrict__ b_ih,     // [H]
    const float* __restrict__ b_hh,     // [H]
    const float* __restrict__ W_out,    // [OUT=1, H]
    const float* __restrict__ b_out,    // [OUT=1]
    float* __restrict__ out_pred,       // [B]
    float* __restrict__ out_h)          // [B, H]
{
    __shared__ float hbuf[ROWS * LSTRIDE];

    const int lane = threadIdx.x;       // 0..31
    const int r0   = blockIdx.x * ROWS; // batch row base for this tile
    const int lm   = lane & 15;         // 0..15
    const int lh   = lane >> 4;         // 0 or 1

    // ---- Weight B-fragments (B = W^T, 4x16 f32 per K-chunk, 2 VGPRs) ----
    // Element (k, n): k = 4c + 2*lh + j, n = nhalf*16 + lm.  B[k][n] = W[n][k].
    v2f bih[2][8], bhh[2][8];
#pragma unroll
    for (int half = 0; half < 2; ++half) {
        const int n = half * 16 + lm;
#pragma unroll
        for (int c = 0; c < 8; ++c) {
            const int k = 4 * c + 2 * lh;
            bih[half][c] = *(const v2f*)(W_ih + n * RNN_IN + k);
            bhh[half][c] = *(const v2f*)(W_hh + n * RNN_H + k);
        }
    }

    // Per-lane bias for C/D layout: element (m,n) sits in lane with n = lm (+16).
    const float bias0 = b_ih[lm] + b_hh[lm];
    const float bias1 = b_ih[16 + lm] + b_hh[16 + lm];

    // ---- h0 -> LDS (natural 16x32 tile, padded rows) ----
    for (int idx = lane; idx < ROWS * RNN_H; idx += 32) {
        const int m = idx >> 5, n = idx & 31;
        hbuf[m * LSTRIDE + n] = hidden[(r0 + m) * RNN_H + n];
    }
    __syncthreads();

    // Each lane's x pointer: row (r0+lm); both lane-halves read the same rows,
    // at k offset 2*lh within each 4-wide chunk. 8-byte aligned v2f loads.
    const float* xrow = x + (size_t)(r0 + lm) * RNN_T * RNN_IN + 2 * lh;

    // Software-pipelined x A-fragments: ax = step t, axn = step t+1.
    v2f ax[8];
#pragma unroll
    for (int c = 0; c < 8; ++c)
        ax[c] = *(const v2f*)(xrow + 4 * c);

    for (int t = 0; t < RNN_T; ++t) {
        // Issue next step's x loads first; they overlap everything below and
        // are only consumed in the next iteration.
        v2f axn[8];
        if (t + 1 < RNN_T) {
            const float* xnext = xrow + (t + 1) * RNN_IN;
            if (t + 2 < RNN_T)
                __builtin_prefetch(xnext + RNN_IN, 0, 1);   // global_prefetch_b8
#pragma unroll
            for (int c = 0; c < 8; ++c)
                axn[c] = *(const v2f*)(xnext + 4 * c);
        }

        // ---- input projection chain (independent of h) ----
        v8f accA, accB;
#pragma unroll
        for (int v = 0; v < 8; ++v) { accA[v] = bias0; accB[v] = bias1; }
#pragma unroll
        for (int c = 0; c < 8; ++c) {
            accA = __builtin_amdgcn_wmma_f32_16x16x4_f32(
                false, ax[c], false, bih[0][c], (short)0, accA, false, false);
            accB = __builtin_amdgcn_wmma_f32_16x16x4_f32(
                false, ax[c], false, bih[1][c], (short)0, accB, false, false);
        }

        // ---- recurrent projection chain (gated by LDS h round-trip) ----
        // A-fragment (16x4 f32): lane holds (m = lm, k = 4c + 2*lh + j).
        v8f hhA = {}, hhB = {};
#pragma unroll
        for (int c = 0; c < 8; ++c) {
            const v2f a = *(const v2f*)(&hbuf[lm * LSTRIDE + 4 * c + 2 * lh]);
            hhA = __builtin_amdgcn_wmma_f32_16x16x4_f32(
                false, a, false, bhh[0][c], (short)0, hhA, false, false);
            hhB = __builtin_amdgcn_wmma_f32_16x16x4_f32(
                false, a, false, bhh[1][c], (short)0, hhB, false, false);
        }

        // ---- combine + relu ----
#pragma unroll
        for (int v = 0; v < 8; ++v) {
            accA[v] = fmaxf(accA[v] + hhA[v], 0.0f);
            accB[v] = fmaxf(accB[v] + hhB[v], 0.0f);
        }

        __syncthreads();  // all lanes done reading h_{t-1} before overwrite

        // ---- D-layout (VGPR v: m = v + 8*lh, n = lm (+16)) back to LDS ----
#pragma unroll
        for (int v = 0; v < 8; ++v) {
            const int m = v + 8 * lh;
            hbuf[m * LSTRIDE + lm]      = accA[v];
            hbuf[m * LSTRIDE + 16 + lm] = accB[v];
        }
        __syncthreads();

        // rotate the x pipeline
#pragma unroll
        for (int c = 0; c < 8; ++c) ax[c] = axn[c];
    }

    // ---- outputs: pred = hT @ W_out^T + b_out (OUT == 1), plus hT itself ----
    if (lane < ROWS) {
        float s = b_out[0];
#pragma unroll
        for (int k = 0; k < RNN_H; ++k)
            s += hbuf[lane * LSTRIDE + k] * W_out[k];
        out_pred[r0 + lane] = s;
    }
    for (int idx = lane; idx < ROWS * RNN_H; idx += 32) {
        const int m = idx >> 5, n = idx & 31;
        out_h[(r0 + m) * RNN_H + n] = hbuf[m * LSTRIDE + n];
    }
}

extern "C" void kernel_launch(void* const* d_in, const int* in_sizes, int n_in,
                              void* d_out, int out_size, void* d_ws, size_t ws_size,
                              hipStream_t stream) {
    (void)in_sizes; (void)n_in; (void)out_size; (void)d_ws; (void)ws_size;
    const float* x      = (const float*)d_in[0];
    const float* hidden = (const float*)d_in[1];
    const float* W_ih   = (const float*)d_in[2];
    const float* W_hh   = (const float*)d_in[3];
    const float* b_ih   = (const float*)d_in[4];
    const float* b_hh   = (const float*)d_in[5];
    const float* W_out  = (const float*)d_in[6];
    const float* b_out  = (const float*)d_in[7];

    float* out       = (float*)d_out;
    float* out_pred  = out;            // [B, 1] — first return value
    float* out_hT    = out + RNN_B;    // [1, B, H] — second return value

    SimpleRNN_28578712387987_kernel<<<RNN_B / ROWS, 32, 0, stream>>>(
        x, hidden, W_ih, W_hh, b_ih, b_hh, W_out, b_out, out_pred, out_hT);
}